// ChebyshevLayer_3315714753165
// MI455X (gfx1250) — compile-verified
//
#include <hip/hip_runtime.h>

typedef __attribute__((ext_vector_type(2))) float v2f;
typedef __attribute__((ext_vector_type(8))) float v8f;

// ---------------------------------------------------------------------------
// T0[m][f*8+n] = x[n][m][f]   (X0 = transpose(x,(1,2,0)).reshape(M, Fin*N))
// ---------------------------------------------------------------------------
__global__ __launch_bounds__(256)
void pack_x(const float* __restrict__ x, float* __restrict__ T0, int M) {
    long idx = (long)blockIdx.x * blockDim.x + threadIdx.x;   // over N*M*FIN
    long total = (long)M * 128;
    if (idx >= total) return;
    int f = (int)(idx & 15);
    long rem = idx >> 4;
    int m = (int)(rem % M);
    int n = (int)(rem / M);
    T0[(size_t)m * 128 + f * 8 + n] = x[idx];
}

__global__ __launch_bounds__(256)
void zero4(float4* __restrict__ p, long n4) {
    long i = (long)blockIdx.x * blockDim.x + threadIdx.x;
    if (i < n4) p[i] = make_float4(0.f, 0.f, 0.f, 0.f);
}

// ---------------------------------------------------------------------------
// Y[row] += val * X[col]  (COO SpMM, 128 columns; 4 cols per thread)
// Hardware f32 atomics (relaxed, agent scope -> global_atomic_add_f32).
// ---------------------------------------------------------------------------
__global__ __launch_bounds__(256)
void spmm_scatter(const float* __restrict__ vals, const int* __restrict__ rows,
                  const int* __restrict__ cols, const float* __restrict__ X,
                  float* __restrict__ Y, int nnz) {
    long t = (long)blockIdx.x * blockDim.x + threadIdx.x;
    long e = t >> 5;                       // 32 threads per nonzero
    if (e >= nnz) return;
    int cg = ((int)t & 31) * 4;            // 4 columns per thread
    float v = vals[e];
    int col = cols[e];
    int row = rows[e];
    const float4 xv = *(const float4*)(X + (size_t)col * 128 + cg);
    float* y = Y + (size_t)row * 128 + cg;
    __hip_atomic_fetch_add(y + 0, v * xv.x, __ATOMIC_RELAXED, __HIP_MEMORY_SCOPE_AGENT);
    __hip_atomic_fetch_add(y + 1, v * xv.y, __ATOMIC_RELAXED, __HIP_MEMORY_SCOPE_AGENT);
    __hip_atomic_fetch_add(y + 2, v * xv.z, __ATOMIC_RELAXED, __HIP_MEMORY_SCOPE_AGENT);
    __hip_atomic_fetch_add(y + 3, v * xv.w, __ATOMIC_RELAXED, __HIP_MEMORY_SCOPE_AGENT);
}

// Chebyshev recurrence epilogue: Tk = 2*Tk - Tprevprev (vectorized float4)
__global__ __launch_bounds__(256)
void combine4(float4* __restrict__ y, const float4* __restrict__ prev, long n4) {
    long i = (long)blockIdx.x * blockDim.x + threadIdx.x;
    if (i >= n4) return;
    float4 a = y[i];
    float4 p = prev[i];
    a.x = 2.f * a.x - p.x;
    a.y = 2.f * a.y - p.y;
    a.z = 2.f * a.z - p.z;
    a.w = 2.f * a.w - p.w;
    y[i] = a;
}

// ---------------------------------------------------------------------------
// Final contraction via V_WMMA_F32_16X16X4_F32:
//   out[n,m,o] = sum_{f,k} T_k[m][f*8+n] * w[f,k,o] + bias[o]
// Rows r = m*8+n (16 per tile), contraction c = f*4+k (16 chunks of K=4),
// two 16x16 output tiles cover FOUT=32. One wave per 16-row tile.
//
// 32-bit WMMA layouts (ISA 7.12.2):
//   A 16x4:  lane tn (half=0) -> VGPR0=A[tn][0], VGPR1=A[tn][1]
//            lane tn (half=1) -> VGPR0=A[tn][2], VGPR1=A[tn][3]
//   B 4x16:  VGPR0 = B[2*half][col=tn], VGPR1 = B[2*half+1][col=tn]
//   D 16x16: VGPR i -> row i + 8*half, col tn
// ---------------------------------------------------------------------------
__global__ __launch_bounds__(256)
void cheb_gemm_wmma(const float* __restrict__ T0, const float* __restrict__ T1,
                    const float* __restrict__ T2, const float* __restrict__ T3,
                    const float* __restrict__ w, const float* __restrict__ bias,
                    float* __restrict__ out, int M) {
    __shared__ float sw[16 * 4 * 32];                 // w: [Fin=16][K=4][Fout=32]
    for (int i = threadIdx.x; i < 2048; i += 256) sw[i] = w[i];
    __syncthreads();

    const int lane = threadIdx.x & 31;
    const int wave = threadIdx.x >> 5;
    const int tile = blockIdx.x * 8 + wave;           // 16-row tiles
    const long totalRows = (long)M * 8;
    if ((long)tile * 16 >= totalRows) return;         // wave-uniform: EXEC stays all-1s

    const int half = lane >> 4;                       // 0: K-pair {0,1}, 1: {2,3}
    const int tn = lane & 15;
    const int r = tile * 16 + tn;                     // this lane's A-row
    const int m = r >> 3;
    const int n = r & 7;
    const float* TA = half ? T2 : T0;                 // chunk c=f*4+k: kk = 2*half+{0,1}
    const float* TB = half ? T3 : T1;
    const size_t abase = (size_t)m * 128 + n;
    const int kk0 = 2 * half;

    v8f acc0 = {0.f, 0.f, 0.f, 0.f, 0.f, 0.f, 0.f, 0.f};
    v8f acc1 = {0.f, 0.f, 0.f, 0.f, 0.f, 0.f, 0.f, 0.f};
#pragma unroll
    for (int f = 0; f < 16; ++f) {
        v2f a;
        a.x = TA[abase + f * 8];                      // A[tn][kk0]   = T_{kk0}[m][f*8+n]
        a.y = TB[abase + f * 8];                      // A[tn][kk0+1]
        v2f b0, b1;
        b0.x = sw[f * 128 + (kk0 + 0) * 32 + tn];     // B[kk0][o]   , o-tile 0
        b0.y = sw[f * 128 + (kk0 + 1) * 32 + tn];     // B[kk0+1][o]
        b1.x = sw[f * 128 + (kk0 + 0) * 32 + 16 + tn];// o-tile 1 (o = 16..31)
        b1.y = sw[f * 128 + (kk0 + 1) * 32 + 16 + tn];
        acc0 = __builtin_amdgcn_wmma_f32_16x16x4_f32(false, a, false, b0,
                                                     (short)0, acc0, false, false);
        acc1 = __builtin_amdgcn_wmma_f32_16x16x4_f32(false, a, false, b1,
                                                     (short)0, acc1, false, false);
    }

    const float bx0 = bias[tn];
    const float bx1 = bias[16 + tn];
#pragma unroll
    for (int i = 0; i < 8; ++i) {
        int rr = tile * 16 + i + 8 * half;            // D row for VGPR i
        int mm = rr >> 3;
        int nn = rr & 7;
        size_t ob = (size_t)nn * (size_t)M * 32 + (size_t)mm * 32;  // out[n,m,o]
        out[ob + tn]      = acc0[i] + bx0;
        out[ob + 16 + tn] = acc1[i] + bx1;
    }
}

// ---------------------------------------------------------------------------
extern "C" void kernel_launch(void* const* d_in, const int* in_sizes, int n_in,
                              void* d_out, int out_size, void* d_ws, size_t ws_size,
                              hipStream_t stream) {
    const float* x      = (const float*)d_in[0];   // [N=8, M, Fin=16]
    const float* l_vals = (const float*)d_in[1];   // [NNZ]
    const float* w      = (const float*)d_in[2];   // [16, 4, 32]
    const float* bias   = (const float*)d_in[3];   // [1, 1, 32]
    const int*   l_row  = (const int*)d_in[4];     // [NNZ]
    const int*   l_col  = (const int*)d_in[5];     // [NNZ]
    float*       out    = (float*)d_out;           // [8, M, 32]

    const int M   = in_sizes[0] / (8 * 16);
    const int NNZ = in_sizes[1];
    const size_t tsize = (size_t)M * 128;          // floats per Chebyshev term

    float* T0 = (float*)d_ws;                      // 4 term buffers (102.4 MB total)
    float* T1 = T0 + tsize;
    float* T2 = T1 + tsize;
    float* T3 = T2 + tsize;

    // zero T1..T3 accumulators
    {
        long n4 = (long)tsize * 3 / 4;
        zero4<<<(int)((n4 + 255) / 256), 256, 0, stream>>>((float4*)T1, n4);
    }
    // T0 = packed/transposed X0
    {
        long n = (long)M * 128;
        pack_x<<<(int)((n + 255) / 256), 256, 0, stream>>>(x, T0, M);
    }
    const long spmmThreads = (long)NNZ * 32;
    const int  spmmBlocks  = (int)((spmmThreads + 255) / 256);
    const long n4   = (long)tsize / 4;
    const int  cb   = (int)((n4 + 255) / 256);

    // T1 = L @ T0
    spmm_scatter<<<spmmBlocks, 256, 0, stream>>>(l_vals, l_row, l_col, T0, T1, NNZ);
    // T2 = 2 * (L @ T1) - T0
    spmm_scatter<<<spmmBlocks, 256, 0, stream>>>(l_vals, l_row, l_col, T1, T2, NNZ);
    combine4<<<cb, 256, 0, stream>>>((float4*)T2, (const float4*)T0, n4);
    // T3 = 2 * (L @ T2) - T1
    spmm_scatter<<<spmmBlocks, 256, 0, stream>>>(l_vals, l_row, l_col, T2, T3, NNZ);
    combine4<<<cb, 256, 0, stream>>>((float4*)T3, (const float4*)T1, n4);

    // out = einsum + bias via f32 WMMA
    const int tiles  = (int)(((long)M * 8 + 15) / 16);
    const int blocks = (tiles + 7) / 8;            // 8 waves (tiles) per 256-thread block
    cheb_gemm_wmma<<<blocks, 256, 0, stream>>>(T0, T1, T2, T3, w, bias, out, M);
}